// SplineCnn_90692529422656
// MI455X (gfx1250) — compile-verified
//
#include <hip/hip_runtime.h>
#include <math.h>

typedef float v2f __attribute__((ext_vector_type(2)));
typedef float v8f __attribute__((ext_vector_type(8)));

#define F_IN 32
#define HID  16
#define NCLS 10

// ---------------- utility: zero a float buffer ----------------
__global__ __launch_bounds__(256) void k_zero(float* __restrict__ p, long n) {
    long i = (long)blockIdx.x * blockDim.x + threadIdx.x;
    if (i < n) p[i] = 0.0f;
}

// ---------------- pack W1[0] | W1[1] | root1 into Wc1 [32][48] ----------------
__global__ __launch_bounds__(256) void k_pack1(const float* __restrict__ W1,   // [2][32][16]
                                               const float* __restrict__ root1,// [32][16]
                                               float* __restrict__ Wc1) {      // [32][48]
    int t = blockIdx.x * blockDim.x + threadIdx.x;
    if (t >= 32 * 48) return;
    int i = t / 48, j = t % 48;
    float v;
    if (j < 16)       v = W1[i * 16 + j];                 // k=0 block
    else if (j < 32)  v = W1[512 + i * 16 + (j - 16)];    // k=1 block
    else              v = root1[i * 16 + (j - 32)];
    Wc1[i * 48 + j] = v;
}

// ---------------- pack W2[0] | W2[1] | root2 into Wc2 [16][32] (cols 30,31 = 0) ----------------
__global__ __launch_bounds__(256) void k_pack2(const float* __restrict__ W2,   // [2][16][10]
                                               const float* __restrict__ root2,// [16][10]
                                               float* __restrict__ Wc2) {      // [16][32]
    int t = blockIdx.x * blockDim.x + threadIdx.x;
    if (t >= 16 * 32) return;
    int i = t / 32, j = t % 32;
    float v = 0.0f;
    if (j < 10)       v = W2[i * 10 + j];
    else if (j < 20)  v = W2[160 + i * 10 + (j - 10)];
    else if (j < 30)  v = root2[i * 10 + (j - 20)];
    Wc2[i * 32 + j] = v;
}

// ---------------- WMMA f32 GEMM: Y[nrows][NT*16] = A[nrows][KDIM] x B[KDIM][NT*16] ----------------
// One wave computes one 16-row tile across all NT column tiles using V_WMMA_F32_16X16X4_F32.
template<int KDIM, int NT>
__global__ __launch_bounds__(256) void k_gemm_wmma(const float* __restrict__ A, int lda,
                                                   const float* __restrict__ B,
                                                   float* __restrict__ Y, int nrows) {
    const int lane = threadIdx.x & 31;
    const int wave = threadIdx.x >> 5;
    const int wavesPerBlk = blockDim.x >> 5;
    const int tile = blockIdx.x * wavesPerBlk + wave;
    const int ntiles = (nrows + 15) >> 4;
    if (tile >= ntiles) return;                 // wave-uniform branch: EXEC stays all-ones

    const int row0 = tile * 16;
    const int r    = lane & 15;                 // A: row-in-tile, B/D: col-in-tile
    const int kh   = lane >> 4;                 // selects K pair / M half per ISA layout
    const int arow = min(row0 + r, nrows - 1);  // clamp (N is a multiple of 16, safety only)

    v8f acc[NT];
#pragma unroll
    for (int t = 0; t < NT; ++t)
#pragma unroll
        for (int i = 0; i < 8; ++i) acc[t][i] = 0.0f;

#pragma unroll
    for (int k0 = 0; k0 < KDIM; k0 += 4) {
        const int ka = k0 + kh * 2;
        v2f a;                                  // A 16x4: lane<16 -> K={k0,k0+1}, lane>=16 -> K={k0+2,k0+3}
        a.x = A[(size_t)arow * lda + ka];
        a.y = A[(size_t)arow * lda + ka + 1];
#pragma unroll
        for (int t = 0; t < NT; ++t) {
            const int n = t * 16 + r;
            v2f b;                              // B 4x16: same K striping, lane%16 = column
            b.x = B[(size_t)ka       * (NT * 16) + n];
            b.y = B[(size_t)(ka + 1) * (NT * 16) + n];
            acc[t] = __builtin_amdgcn_wmma_f32_16x16x4_f32(
                false, a, false, b, (short)0, acc[t], false, false);
        }
    }

    // D 16x16 f32: VGPR i, lanes 0-15 -> M=i, lanes 16-31 -> M=8+i; N = lane%16
#pragma unroll
    for (int t = 0; t < NT; ++t)
#pragma unroll
        for (int i = 0; i < 8; ++i) {
            const int m = row0 + i + 8 * kh;
            if (m < nrows)
                Y[(size_t)m * (NT * 16) + t * 16 + r] = acc[t][i];
        }
}

// ---------------- edge pass layer 1: msg = b0*y0[src] + b1*y1[src]; scatter-mean ----------------
__global__ __launch_bounds__(256) void k_edge1(const long long* __restrict__ ei,   // [2][E]
                                               const float* __restrict__ attr,     // [E]
                                               const float* __restrict__ Y,        // [N][48]
                                               float* __restrict__ agg,            // [N][16]
                                               float* __restrict__ deg,            // [N]
                                               int E) {
    long gid = (long)blockIdx.x * blockDim.x + threadIdx.x;
    if (gid >= (long)E * 16) return;
    const int e = (int)(gid >> 4);
    const int c = (int)(gid & 15);
    const int src = (int)ei[e];
    const int dst = (int)ei[(long)E + e];
    const float v  = attr[e];
    const float b0 = fmaxf(0.0f, 1.0f - fabsf(v));
    const float b1 = fmaxf(0.0f, 1.0f - fabsf(v - 1.0f));
    const float m  = b0 * Y[(size_t)src * 48 + c] + b1 * Y[(size_t)src * 48 + 16 + c];
    atomicAdd(&agg[(size_t)dst * 16 + c], m);
    if (c == 0) atomicAdd(&deg[dst], 1.0f);
}

// ---------------- node pass: h = elu(agg/max(deg,1) + x@root1 + b1) ----------------
__global__ __launch_bounds__(256) void k_hidden(const float* __restrict__ agg,  // [N][16]
                                                const float* __restrict__ Y,    // [N][48] (cols 32..47 = x@root1)
                                                const float* __restrict__ b1,   // [16]
                                                const float* __restrict__ deg,  // [N]
                                                float* __restrict__ h,          // [N][16]
                                                int Nn) {
    long idx = (long)blockIdx.x * blockDim.x + threadIdx.x;
    if (idx >= (long)Nn * 16) return;
    const int n = (int)(idx >> 4);
    const int c = (int)(idx & 15);
    const float d   = fmaxf(deg[n], 1.0f);
    const float val = agg[idx] / d + Y[(size_t)n * 48 + 32 + c] + b1[c];
    h[idx] = val > 0.0f ? val : expm1f(val);
}

// ---------------- edge pass layer 2 ----------------
__global__ __launch_bounds__(256) void k_edge2(const long long* __restrict__ ei,
                                               const float* __restrict__ attr,
                                               const float* __restrict__ Z,    // [N][32]: z0|z1|hr|pad
                                               float* __restrict__ agg,        // [N][16]
                                               int E) {
    long gid = (long)blockIdx.x * blockDim.x + threadIdx.x;
    if (gid >= (long)E * 16) return;
    const int e = (int)(gid >> 4);
    const int c = (int)(gid & 15);
    if (c >= NCLS) return;
    const int src = (int)ei[e];
    const int dst = (int)ei[(long)E + e];
    const float v  = attr[e];
    const float b0 = fmaxf(0.0f, 1.0f - fabsf(v));
    const float b1 = fmaxf(0.0f, 1.0f - fabsf(v - 1.0f));
    const float m  = b0 * Z[(size_t)src * 32 + c] + b1 * Z[(size_t)src * 32 + 10 + c];
    atomicAdd(&agg[(size_t)dst * 16 + c], m);
}

// ---------------- final: out = log_softmax(agg/max(deg,1) + h@root2 + b2) ----------------
__global__ __launch_bounds__(256) void k_final(const float* __restrict__ agg,  // [N][16]
                                               const float* __restrict__ Z,    // [N][32] (cols 20..29 = h@root2)
                                               const float* __restrict__ b2,   // [10]
                                               const float* __restrict__ deg,  // [N]
                                               float* __restrict__ out,        // [N][10]
                                               int Nn) {
    int n = blockIdx.x * blockDim.x + threadIdx.x;
    if (n >= Nn) return;
    const float d = fmaxf(deg[n], 1.0f);
    float vals[NCLS];
    float mx = -1e30f;
#pragma unroll
    for (int c = 0; c < NCLS; ++c) {
        vals[c] = agg[(size_t)n * 16 + c] / d + Z[(size_t)n * 32 + 20 + c] + b2[c];
        mx = fmaxf(mx, vals[c]);
    }
    float s = 0.0f;
#pragma unroll
    for (int c = 0; c < NCLS; ++c) s += __expf(vals[c] - mx);
    const float lse = mx + logf(s);
#pragma unroll
    for (int c = 0; c < NCLS; ++c) out[(size_t)n * 10 + c] = vals[c] - lse;
}

extern "C" void kernel_launch(void* const* d_in, const int* in_sizes, int n_in,
                              void* d_out, int out_size, void* d_ws, size_t ws_size,
                              hipStream_t stream) {
    const float*     x     = (const float*)d_in[0];      // [N,32]
    const long long* ei    = (const long long*)d_in[1];  // [2,E] int64
    const float*     attr  = (const float*)d_in[2];      // [E,1]
    const float*     W1    = (const float*)d_in[3];      // [2,32,16]
    const float*     root1 = (const float*)d_in[4];      // [32,16]
    const float*     b1    = (const float*)d_in[5];      // [16]
    const float*     W2    = (const float*)d_in[6];      // [2,16,10]
    const float*     root2 = (const float*)d_in[7];      // [16,10]
    const float*     b2    = (const float*)d_in[8];      // [10]
    float*           out   = (float*)d_out;

    const int Nn = in_sizes[0] / F_IN;     // 100000
    const int E  = in_sizes[1] / 2;        // 1600000

    // workspace carving (floats)
    float* ws = (float*)d_ws;
    size_t o = 0;
    float* Y    = ws + o; o += (size_t)Nn * 48;   // y0 | y1 | x@root1
    float* Z    = ws + o; o += (size_t)Nn * 32;   // z0 | z1 | h@root2 | pad
    float* agg1 = ws + o; o += (size_t)Nn * 16;
    float* agg2 = ws + o; o += (size_t)Nn * 16;
    float* deg  = ws + o; o += (size_t)Nn;
    float* h    = ws + o; o += (size_t)Nn * 16;
    float* Wc1  = ws + o; o += 32 * 48;
    float* Wc2  = ws + o; o += 16 * 32;

    // 1) pack weights
    k_pack1<<<(32 * 48 + 255) / 256, 256, 0, stream>>>(W1, root1, Wc1);
    k_pack2<<<(16 * 32 + 255) / 256, 256, 0, stream>>>(W2, root2, Wc2);

    // 2) zero accumulators
    k_zero<<<(int)(((long)Nn * 16 + 255) / 256), 256, 0, stream>>>(agg1, (long)Nn * 16);
    k_zero<<<(int)(((long)Nn * 16 + 255) / 256), 256, 0, stream>>>(agg2, (long)Nn * 16);
    k_zero<<<(Nn + 255) / 256, 256, 0, stream>>>(deg, Nn);

    // 3) layer-1 node GEMM (WMMA f32): Y = x @ [W1_0 | W1_1 | root1]
    {
        const int tiles = (Nn + 15) / 16;
        const int blocks = (tiles + 7) / 8;       // 8 waves / 256-thread block
        k_gemm_wmma<F_IN, 3><<<blocks, 256, 0, stream>>>(x, F_IN, Wc1, Y, Nn);
    }

    // 4) layer-1 edge pass (gather + blend + scatter-add + degree)
    k_edge1<<<(int)(((long)E * 16 + 255) / 256), 256, 0, stream>>>(ei, attr, Y, agg1, deg, E);

    // 5) hidden activation
    k_hidden<<<(int)(((long)Nn * 16 + 255) / 256), 256, 0, stream>>>(agg1, Y, b1, deg, h, Nn);

    // 6) layer-2 node GEMM (WMMA f32): Z = h @ [W2_0 | W2_1 | root2 | 0]
    {
        const int tiles = (Nn + 15) / 16;
        const int blocks = (tiles + 7) / 8;
        k_gemm_wmma<HID, 2><<<blocks, 256, 0, stream>>>(h, HID, Wc2, Z, Nn);
    }

    // 7) layer-2 edge pass
    k_edge2<<<(int)(((long)E * 16 + 255) / 256), 256, 0, stream>>>(ei, attr, Z, agg2, E);

    // 8) mean + root + bias + log_softmax
    k_final<<<(Nn + 255) / 256, 256, 0, stream>>>(agg2, Z, b2, deg, out, Nn);
}